// PointWarper_73108933312574
// MI455X (gfx1250) — compile-verified
//
#include <hip/hip_runtime.h>
#include <math.h>

typedef __attribute__((ext_vector_type(2))) float v2f;
typedef __attribute__((ext_vector_type(8))) float v8f;

#define J_BONES 24
#define HDIM 256
#define TDIM 32
#define NPARAMS 100   // (J+1)*4

// ---------------------------------------------------------------------------
// Kernel 1: the tiny MLP on t (single block, 256 threads). Writes 100 params
// to ws[0..99].
// ---------------------------------------------------------------------------
__global__ __launch_bounds__(256) void mlp_kernel(
    const float* __restrict__ t,
    const float* __restrict__ W0, const float* __restrict__ b0,
    const float* __restrict__ W1, const float* __restrict__ b1,
    const float* __restrict__ W2, const float* __restrict__ b2,
    const float* __restrict__ W3, const float* __restrict__ b3,
    const float* __restrict__ W4,
    float* __restrict__ params)
{
    __shared__ float bufA[HDIM];
    __shared__ float bufB[HDIM];
    __shared__ float tl[TDIM];
    const int i = threadIdx.x;

    if (i < TDIM) tl[i] = t[i];
    __syncthreads();

    // layer 0: (32 -> 256)
    float acc = b0[i];
    #pragma unroll 4
    for (int k = 0; k < TDIM; ++k) acc += tl[k] * W0[k * HDIM + i];
    bufA[i] = fmaxf(acc, 0.0f);
    __syncthreads();

    // layer 1
    acc = b1[i];
    #pragma unroll 4
    for (int k = 0; k < HDIM; ++k) acc += bufA[k] * W1[k * HDIM + i];
    bufB[i] = fmaxf(acc, 0.0f);
    __syncthreads();

    // layer 2
    acc = b2[i];
    #pragma unroll 4
    for (int k = 0; k < HDIM; ++k) acc += bufB[k] * W2[k * HDIM + i];
    bufA[i] = fmaxf(acc, 0.0f);
    __syncthreads();

    // layer 3
    acc = b3[i];
    #pragma unroll 4
    for (int k = 0; k < HDIM; ++k) acc += bufA[k] * W3[k * HDIM + i];
    bufB[i] = fmaxf(acc, 0.0f);
    __syncthreads();

    // output layer: (256 -> 100), no bias, no relu
    if (i < NPARAMS) {
        float o = 0.0f;
        #pragma unroll 4
        for (int k = 0; k < HDIM; ++k) o += bufB[k] * W4[k * NPARAMS + i];
        params[i] = o;
    }
}

// ---------------------------------------------------------------------------
// Kernel 2: Rodrigues + bone transforms + tree chain products (1 wave).
//  - writes Bmat[24][16] (rows 0..2 of each global bone transform, flattened
//    row-major into 12 cols, cols 12..15 zero) into ws
//  - writes joints_warped_rel (24*3 floats) into d_out tail
// ---------------------------------------------------------------------------
__global__ __launch_bounds__(32) void bones_kernel(
    const float* __restrict__ params,       // 100 floats (from kernel 1)
    const float* __restrict__ joints,       // 24*3
    const int*   __restrict__ parent_indices, // 24*depth
    const int*   __restrict__ parent_joint_ex, // 24
    int depth,
    float* __restrict__ Bmat,               // ws: 24*16 floats
    float* __restrict__ out_joints)         // d_out + N*3
{
    __shared__ float M[J_BONES + 1][16];    // M[0] = identity, M[j+1] = bone j
    const int j = threadIdx.x;

    if (j == J_BONES) {
        #pragma unroll
        for (int c = 0; c < 16; ++c) M[0][c] = (c % 5 == 0) ? 1.0f : 0.0f;
    }
    if (j < J_BONES) {
        const float px = params[j * 4 + 0];
        const float py = params[j * 4 + 1];
        const float pz = params[j * 4 + 2];
        const float th = params[j * 4 + 3];
        const float inv = 1.0f / sqrtf(1e-5f + px * px + py * py + pz * pz);
        const float x = px * inv, y = py * inv, z = pz * inv;
        const float c = cosf(th), s = sinf(th);
        const float omc = 1.0f - c;
        float R[9];
        R[0] = x * x + (1.0f - x * x) * c;
        R[1] = x * y * omc - z * s;
        R[2] = x * z * omc + y * s;
        R[3] = x * y * omc + z * s;
        R[4] = y * y + (1.0f - y * y) * c;
        R[5] = y * z * omc - x * s;
        R[6] = x * z * omc - y * s;
        R[7] = y * z * omc + x * s;
        R[8] = z * z + (1.0f - z * z) * c;

        const int pj = parent_joint_ex[j];
        float vx = 0.0f, vy = 0.0f, vz = 0.0f;
        if (pj >= 0) { vx = joints[pj * 3 + 0]; vy = joints[pj * 3 + 1]; vz = joints[pj * 3 + 2]; }
        const float tx = vx - (R[0] * vx + R[1] * vy + R[2] * vz);
        const float ty = vy - (R[3] * vx + R[4] * vy + R[5] * vz);
        const float tz = vz - (R[6] * vx + R[7] * vy + R[8] * vz);

        float* Mj = M[j + 1];
        Mj[0] = R[0]; Mj[1]  = R[1]; Mj[2]  = R[2]; Mj[3]  = tx;
        Mj[4] = R[3]; Mj[5]  = R[4]; Mj[6]  = R[5]; Mj[7]  = ty;
        Mj[8] = R[6]; Mj[9]  = R[7]; Mj[10] = R[8]; Mj[11] = tz;
        Mj[12] = 0.0f; Mj[13] = 0.0f; Mj[14] = 0.0f; Mj[15] = 1.0f;
    }
    __syncthreads();

    if (j < J_BONES) {
        // path product, root first: T = M[p0] @ M[p1] @ ... ; pi==-1 -> identity
        float T[16] = {1,0,0,0, 0,1,0,0, 0,0,1,0, 0,0,0,1};
        for (int d = 0; d < depth; ++d) {
            const int idx = parent_indices[j * depth + d] + 1; // -1 -> 0 (identity)
            const float* Md = M[idx];
            float Tn[16];
            #pragma unroll
            for (int r = 0; r < 4; ++r) {
                #pragma unroll
                for (int cc = 0; cc < 4; ++cc) {
                    float a = T[r * 4 + 0] * Md[0 * 4 + cc];
                    a += T[r * 4 + 1] * Md[1 * 4 + cc];
                    a += T[r * 4 + 2] * Md[2 * 4 + cc];
                    a += T[r * 4 + 3] * Md[3 * 4 + cc];
                    Tn[r * 4 + cc] = a;
                }
            }
            #pragma unroll
            for (int q = 0; q < 16; ++q) T[q] = Tn[q];
        }
        // B matrix for the WMMA blend: rows 0..2 flattened (12 cols), pad to 16
        #pragma unroll
        for (int cc = 0; cc < 12; ++cc) Bmat[j * 16 + cc] = T[cc];
        #pragma unroll
        for (int cc = 12; cc < 16; ++cc) Bmat[j * 16 + cc] = 0.0f;

        // warped joints (relative): (T @ [joint,1])[:3]
        const float jx = joints[j * 3 + 0], jy = joints[j * 3 + 1], jz = joints[j * 3 + 2];
        out_joints[j * 3 + 0] = T[0] * jx + T[1] * jy + T[2]  * jz + T[3];
        out_joints[j * 3 + 1] = T[4] * jx + T[5] * jy + T[6]  * jz + T[7];
        out_joints[j * 3 + 2] = T[8] * jx + T[9] * jy + T[10] * jz + T[11];
    }
}

// ---------------------------------------------------------------------------
// Kernel 3: the hot loop. Blend 24 bone transforms per point with
// V_WMMA_F32_16X16X4_F32 (full fp32), then apply to [x,y,z,1] + global_t.
// Each wave: 2 A-tiles (32 points), shared B fragments, 6 K-steps -> 12 WMMAs.
// Block = 256 threads = 8 waves = 256 points.
// ---------------------------------------------------------------------------
__global__ __launch_bounds__(256) void warp_points_kernel(
    const float* __restrict__ weights,   // N x 24
    const float* __restrict__ pcd,       // N x 3
    const float* __restrict__ Bmat,      // 24 x 16 (ws)
    const float* __restrict__ params,    // global_t = params[96..98]
    float* __restrict__ out,             // N x 3
    int n)
{
    // 16 tiles (8 waves * 2), each 16 rows x 17 floats (padded vs bank conflicts)
    __shared__ float lds[16 * 272];

    const int tid  = threadIdx.x;
    const int wave = tid >> 5;
    const int lane = tid & 31;
    const int m    = lane & 15;          // A row / B col within tile
    const int koff = (lane >> 4) * 2;    // K sub-offset per half-wave

    const long long blockBase = (long long)blockIdx.x * 256;
    long long p0 = blockBase + (long long)wave * 32 + m;   // tile 0 point
    long long p1 = p0 + 16;                                // tile 1 point
    // WMMA needs EXEC all-1s: clamp loads, guard stores later.
    const long long q0 = (p0 < n) ? p0 : (long long)(n - 1);
    const long long q1 = (p1 < n) ? p1 : (long long)(n - 1);

    v8f c0 = {};
    v8f c1 = {};
    #pragma unroll
    for (int s = 0; s < 6; ++s) {
        const int kb = 4 * s;
        // A fragments: contiguous aligned float2 of this point's weights
        v2f a0 = *(const v2f*)(weights + q0 * 24 + kb + koff);
        v2f a1 = *(const v2f*)(weights + q1 * 24 + kb + koff);
        // B fragment: rows (kb+koff, kb+koff+1), column m
        v2f b;
        b.x = Bmat[(kb + koff + 0) * 16 + m];
        b.y = Bmat[(kb + koff + 1) * 16 + m];
        c0 = __builtin_amdgcn_wmma_f32_16x16x4_f32(false, a0, false, b,
                                                   (short)0, c0, false, false);
        c1 = __builtin_amdgcn_wmma_f32_16x16x4_f32(false, a1, false, b,
                                                   (short)0, c1, false, false);
    }

    // Spill D tiles to LDS so each thread can gather its own point's 12 coeffs.
    float* tile0 = &lds[(wave * 2 + 0) * 272];
    float* tile1 = &lds[(wave * 2 + 1) * 272];
    const int rbase = (lane >> 4) * 8;
    const int col   = lane & 15;
    #pragma unroll
    for (int v = 0; v < 8; ++v) {
        tile0[(rbase + v) * 17 + col] = c0[v];
        tile1[(rbase + v) * 17 + col] = c1[v];
    }
    __syncthreads();

    // Epilogue: one point per thread
    const long long p = blockBase + tid;
    if (p < n) {
        const int  tile = tid >> 4;
        const int  row  = tid & 15;
        const float* g  = &lds[tile * 272 + row * 17];
        const float x = pcd[p * 3 + 0];
        const float y = pcd[p * 3 + 1];
        const float z = pcd[p * 3 + 2];
        const float gx = params[96], gy = params[97], gz = params[98];
        out[p * 3 + 0] = g[0] * x + g[1] * y + g[2]  * z + g[3]  + gx;
        out[p * 3 + 1] = g[4] * x + g[5] * y + g[6]  * z + g[7]  + gy;
        out[p * 3 + 2] = g[8] * x + g[9] * y + g[10] * z + g[11] + gz;
    }
}

// ---------------------------------------------------------------------------
// Launch
// ---------------------------------------------------------------------------
extern "C" void kernel_launch(void* const* d_in, const int* in_sizes, int n_in,
                              void* d_out, int out_size, void* d_ws, size_t ws_size,
                              hipStream_t stream)
{
    (void)n_in; (void)out_size; (void)ws_size;
    const float* weights = (const float*)d_in[0];
    const float* joints  = (const float*)d_in[1];
    const float* t       = (const float*)d_in[2];
    const float* pcd     = (const float*)d_in[3];
    const float* W0      = (const float*)d_in[4];
    const float* b0      = (const float*)d_in[5];
    const float* W1      = (const float*)d_in[6];
    const float* b1      = (const float*)d_in[7];
    const float* W2      = (const float*)d_in[8];
    const float* b2      = (const float*)d_in[9];
    const float* W3      = (const float*)d_in[10];
    const float* b3      = (const float*)d_in[11];
    const float* W4      = (const float*)d_in[12];
    const int*   pidx    = (const int*)d_in[13];
    const int*   pjex    = (const int*)d_in[14];

    const int N     = in_sizes[3] / 3;          // canonical_pcd is N x 3
    const int depth = in_sizes[13] / J_BONES;   // parent_indices is J x depth

    float* ws_params = (float*)d_ws;            // 100 floats
    float* ws_Bmat   = (float*)d_ws + 128;      // 24*16 floats
    float* out_xyz   = (float*)d_out;           // N*3
    float* out_j     = (float*)d_out + (size_t)N * 3; // J*3

    mlp_kernel<<<1, 256, 0, stream>>>(t, W0, b0, W1, b1, W2, b2, W3, b3, W4,
                                      ws_params);
    bones_kernel<<<1, 32, 0, stream>>>(ws_params, joints, pidx, pjex, depth,
                                       ws_Bmat, out_j);
    const int nblocks = (N + 255) / 256;
    warp_points_kernel<<<nblocks, 256, 0, stream>>>(weights, pcd, ws_Bmat,
                                                    ws_params, out_xyz, N);
}